// GlobalAttentionGeneral_1271310319860
// MI455X (gfx1250) — compile-verified
//
#include <hip/hip_runtime.h>
#include <hip/hip_bf16.h>
#include <math.h>

// Sizes from the reference
#define B_   32
#define IDF  64
#define RES  128
#define QL   (RES * RES)   // 16384
#define CDF  768
#define SL   18
#define SLP  32            // SL padded to 32 for two 16-wide WMMA N tiles
#define ADIM 100

typedef __attribute__((ext_vector_type(2))) float v2f;
typedef __attribute__((ext_vector_type(8))) float v8f;

// ---------------------------------------------------------------------------
// Kernel 1: sourceT[b,i,s] = sum_c conv_ctx_w[i,c] * context[b,c,s]
//           (stored padded to s=32 with zeros, laid out [b][i][32])
//           cw[b,s] = sum_i conv_w[i] * sourceT[b,i,s]
// ---------------------------------------------------------------------------
__global__ __launch_bounds__(256) void prep_kernel(
    const float* __restrict__ context,     // [B, CDF, SL]
    const float* __restrict__ conv_ctx_w,  // [IDF, CDF]
    const float* __restrict__ conv_w,      // [IDF]
    float* __restrict__ srcT,              // [B, IDF, SLP]
    float* __restrict__ cw)                // [B, SLP]
{
    const int b = blockIdx.x;
    const int tid = threadIdx.x;
    __shared__ float sS[IDF * SLP];

    for (int idx = tid; idx < IDF * SLP; idx += 256) sS[idx] = 0.0f;
    __syncthreads();

    for (int idx = tid; idx < IDF * SL; idx += 256) {
        const int i = idx / SL;
        const int s = idx % SL;
        const float* ctx = context + (size_t)b * CDF * SL + s;
        const float* w   = conv_ctx_w + (size_t)i * CDF;
        float acc = 0.0f;
        for (int c = 0; c < CDF; ++c) acc = fmaf(w[c], ctx[(size_t)c * SL], acc);
        sS[i * SLP + s] = acc;
    }
    __syncthreads();

    for (int idx = tid; idx < IDF * SLP; idx += 256)
        srcT[(size_t)b * IDF * SLP + idx] = sS[idx];

    if (tid < SLP) {
        float acc = 0.0f;
        for (int i = 0; i < IDF; ++i) acc = fmaf(conv_w[i], sS[i * SLP + tid], acc);
        cw[b * SLP + tid] = acc;   // padded cols are exactly zero
    }
}

// ---------------------------------------------------------------------------
// Kernel 2 (main): per wave, a 16q x 32s logits tile via V_WMMA_F32_16X16X4_F32
// chained over K=64 (16 steps of K=4), then softmax(s<18) and dot with cw.
//   logits[q,s] = sum_i target[b,i,q] * srcT[b,i,s]     (M=q, N=s, K=i)
// A layout (f32 16x4): lanes 0-15 hold M=lane; VGPR0 = K even of pair,
//   VGPR1 = K odd; lane halves take K pairs (0,1) and (2,3).
// B layout mirrors: VGPR0/1 rows K0..K3, N = lane&15.
// ---------------------------------------------------------------------------
__global__ __launch_bounds__(256) void attn_main_kernel(
    const float* __restrict__ target,  // inputs viewed as [B, IDF, QL]
    const float* __restrict__ srcT,    // [B, IDF, SLP]
    const float* __restrict__ cw,      // [B, SLP]
    const float* __restrict__ conv_b,  // [1]
    float* __restrict__ wctx1)         // [B, QL]
{
    __shared__ float sB[IDF][SLP];         // 8 KB   B-matrix tile (i x s)
    __shared__ float sCw[SLP];
    __shared__ float sLog[8][16][SLP + 2]; // ~17 KB logits, padded vs bank conflicts

    const int b     = blockIdx.y;
    const int chunk = blockIdx.x;          // 128 chunks of 128 q
    const int tid   = threadIdx.x;

    for (int idx = tid; idx < IDF * SLP; idx += 256)
        sB[idx / SLP][idx % SLP] = srcT[(size_t)b * IDF * SLP + idx];
    if (tid < SLP) sCw[tid] = cw[b * SLP + tid];
    __syncthreads();

    const int wave = tid >> 5;
    const int lane = tid & 31;
    const int half = lane >> 4;   // which K pair of the 4-wide step
    const int n    = lane & 15;   // M index for A loads, N index for B/C
    const int Q    = chunk * 128 + wave * 16;
    const float* tgt = target + (size_t)b * IDF * QL;

    v8f c0 = {};  // s = 0..15
    v8f c1 = {};  // s = 16..31 (cols >=18 multiply zeros)
#pragma unroll
    for (int kk = 0; kk < 16; ++kk) {
        const int k0 = kk * 4 + 2 * half;
        v2f a, b0, b1;
        a.x  = tgt[(size_t)(k0 + 0) * QL + Q + n];
        a.y  = tgt[(size_t)(k0 + 1) * QL + Q + n];
        b0.x = sB[k0 + 0][n];       b0.y = sB[k0 + 1][n];
        b1.x = sB[k0 + 0][16 + n];  b1.y = sB[k0 + 1][16 + n];
        c0 = __builtin_amdgcn_wmma_f32_16x16x4_f32(false, a, false, b0,
                                                   (short)0, c0, false, false);
        c1 = __builtin_amdgcn_wmma_f32_16x16x4_f32(false, a, false, b1,
                                                   (short)0, c1, false, false);
    }

    // C layout: VGPR r holds row M=r (lanes 0-15) / M=r+8 (lanes 16-31), col N=n
#pragma unroll
    for (int r = 0; r < 8; ++r) {
        const int row = r + half * 8;
        sLog[wave][row][n]      = c0[r];
        sLog[wave][row][16 + n] = c1[r];
    }
    __syncthreads();

    if (lane < 16) {
        const int row = lane;
        float l[SL];
        float m = -1e30f;
#pragma unroll
        for (int s = 0; s < SL; ++s) { l[s] = sLog[wave][row][s]; m = fmaxf(m, l[s]); }
        float sum = 0.0f;
#pragma unroll
        for (int s = 0; s < SL; ++s) { l[s] = __expf(l[s] - m); sum += l[s]; }
        const float inv = 1.0f / sum;
        float acc = 0.0f;
#pragma unroll
        for (int s = 0; s < SL; ++s) acc = fmaf(l[s], sCw[s], acc);
        wctx1[(size_t)b * QL + Q + row] = acc * inv + conv_b[0];
    }
}

// ---------------------------------------------------------------------------
// Kernel 3: out[b,a] = sum_q wctx1[b,q] * fc_w[a,q] + fc_b[a]
// One block per a: fc_w row read once (64 KB), wctx1 (2 MB) stays in L2.
// ---------------------------------------------------------------------------
__global__ __launch_bounds__(256) void fc_kernel(
    const float* __restrict__ wctx1,  // [B, QL]
    const float* __restrict__ fc_w,   // [ADIM, QL]
    const float* __restrict__ fc_b,   // [ADIM]
    float* __restrict__ out)          // [B, ADIM]
{
    const int a = blockIdx.x;
    const int tid = threadIdx.x;
    __shared__ float red[256];
    const float* wrow = fc_w + (size_t)a * QL;

    for (int b = 0; b < B_; ++b) {
        const float* v = wctx1 + (size_t)b * QL;
        float acc = 0.0f;
        for (int q = tid; q < QL; q += 256) acc = fmaf(wrow[q], v[q], acc);
        red[tid] = acc;
        __syncthreads();
        for (int st = 128; st > 0; st >>= 1) {
            if (tid < st) red[tid] += red[tid + st];
            __syncthreads();
        }
        if (tid == 0) out[b * ADIM + a] = red[0] + fc_b[a];
        __syncthreads();
    }
}

// ---------------------------------------------------------------------------
extern "C" void kernel_launch(void* const* d_in, const int* in_sizes, int n_in,
                              void* d_out, int out_size, void* d_ws, size_t ws_size,
                              hipStream_t stream) {
    const float* inputs     = (const float*)d_in[0]; // [B, IDF, RES, RES]
    const float* context    = (const float*)d_in[1]; // [B, CDF, SL]
    const float* conv_ctx_w = (const float*)d_in[2]; // [IDF, CDF]
    const float* conv_w     = (const float*)d_in[3]; // [IDF]
    const float* conv_b     = (const float*)d_in[4]; // [1]
    const float* fc_w       = (const float*)d_in[5]; // [ADIM, QL]
    const float* fc_b       = (const float*)d_in[6]; // [ADIM]
    float* out = (float*)d_out;                      // [B, ADIM]

    float* srcT  = (float*)d_ws;                      // B*IDF*SLP  = 65536 f
    float* cw    = srcT + (size_t)B_ * IDF * SLP;     // B*SLP      = 1024 f
    float* wctx1 = cw + (size_t)B_ * SLP;             // B*QL       = 524288 f
    (void)in_sizes; (void)n_in; (void)out_size; (void)ws_size;

    prep_kernel<<<dim3(B_), dim3(256), 0, stream>>>(context, conv_ctx_w, conv_w,
                                                    srcT, cw);
    attn_main_kernel<<<dim3(QL / 128, B_), dim3(256), 0, stream>>>(inputs, srcT, cw,
                                                                   conv_b, wctx1);
    fc_kernel<<<dim3(ADIM), dim3(256), 0, stream>>>(wctx1, fc_w, fc_b, out);
}